// Model_69776038691096
// MI455X (gfx1250) — compile-verified
//
#include <hip/hip_runtime.h>
#include <stdint.h>

// Problem geometry from the reference: image (16, 3, 1024, 1024) fp32.
#define NCH    48           // B*C channels
#define NPIX   (1u << 20)   // pixels per channel (1024*1024)
#define NBINS  256

typedef __attribute__((ext_vector_type(4))) float f4;

__device__ __forceinline__ int qbin(float x) {
    // matches jnp: (clip(x,0,1) * 255).astype(int32)  (truncation, x >= 0)
    x = fminf(fmaxf(x, 0.0f), 1.0f);
    return (int)(x * 255.0f);
}

// ---- CDNA5 async global->LDS copy (ASYNCcnt-tracked, CDNA5 ISA) ----
__device__ __forceinline__ void async_copy_b128(const f4* gsrc, f4* lds_dst) {
    // Flat address of a __shared__ object: high 32 bits = LDS aperture,
    // low 32 bits = wave-relative LDS byte offset (ISA §10.2).
    unsigned loff = (unsigned)(unsigned long long)(uintptr_t)lds_dst;
    asm volatile("global_load_async_to_lds_b128 %0, %1, off"
                 :: "v"(loff), "v"(gsrc) : "memory");
}
__device__ __forceinline__ void async_wait_le3() {
    asm volatile("s_wait_asynccnt 0x3" ::: "memory");
}
__device__ __forceinline__ void async_wait_le2() {
    asm volatile("s_wait_asynccnt 0x2" ::: "memory");
}
__device__ __forceinline__ void async_wait_le1() {
    asm volatile("s_wait_asynccnt 0x1" ::: "memory");
}
__device__ __forceinline__ void async_wait_le0() {
    asm volatile("s_wait_asynccnt 0x0" ::: "memory");
}

// -------------------- pass 0: zero the global histograms --------------------
__global__ void __launch_bounds__(256) hist_init(unsigned* __restrict__ ghist) {
    ghist[blockIdx.x * NBINS + threadIdx.x] = 0u;
}

// -------------------- pass 1: per-channel histogram --------------------
// 64 blocks per channel, 256 threads, 16 chunks of 256 lanes x 16B = 4 KB
// (1024 pixels) each => 16384 pixels per block. 4-deep async pipeline keeps
// 3 chunks (1.5 KB/wave) in flight against HBM latency.
__global__ void __launch_bounds__(256) hist_kernel(const f4* __restrict__ img,
                                                   unsigned* __restrict__ ghist) {
    __shared__ unsigned lhist[8][NBINS];   // one sub-histogram per wave32
    __shared__ f4 stage[4][256];           // async 4-deep ring buffer

    const int t = threadIdx.x;
    const int w = t >> 5;                  // wave id (wave32)

#pragma unroll
    for (int j = 0; j < 8; ++j) lhist[j][t] = 0u;
    __syncthreads();

    const int ch  = blockIdx.x >> 6;       // 64 blocks per channel
    const int blk = blockIdx.x & 63;
    const f4* base = img + ((size_t)ch << 18)            // NPIX/4 f4 per channel
                         + ((size_t)blk * (16 * 256));   // this block's chunk

    // prologue: stage chunks 0..2
    async_copy_b128(base + 0 * 256 + t, &stage[0][t]);
    async_copy_b128(base + 1 * 256 + t, &stage[1][t]);
    async_copy_b128(base + 2 * 256 + t, &stage[2][t]);

    // Each lane consumes only its own staged 16B -> wave-local; asynccnt
    // waits are the only sync needed (no barriers in the hot loop).
#pragma unroll 1
    for (int it = 0; it < 13; ++it) {
        async_copy_b128(base + (size_t)(it + 3) * 256 + t, &stage[(it + 3) & 3][t]);
        async_wait_le3();                  // chunk `it` has landed
        f4 v = stage[it & 3][t];
        atomicAdd(&lhist[w][qbin(v.x)], 1u);
        atomicAdd(&lhist[w][qbin(v.y)], 1u);
        atomicAdd(&lhist[w][qbin(v.z)], 1u);
        atomicAdd(&lhist[w][qbin(v.w)], 1u);
    }
    // tail: chunks 13, 14, 15
    {
        async_wait_le2();
        f4 v = stage[13 & 3][t];
        atomicAdd(&lhist[w][qbin(v.x)], 1u);
        atomicAdd(&lhist[w][qbin(v.y)], 1u);
        atomicAdd(&lhist[w][qbin(v.z)], 1u);
        atomicAdd(&lhist[w][qbin(v.w)], 1u);
    }
    {
        async_wait_le1();
        f4 v = stage[14 & 3][t];
        atomicAdd(&lhist[w][qbin(v.x)], 1u);
        atomicAdd(&lhist[w][qbin(v.y)], 1u);
        atomicAdd(&lhist[w][qbin(v.z)], 1u);
        atomicAdd(&lhist[w][qbin(v.w)], 1u);
    }
    {
        async_wait_le0();
        f4 v = stage[15 & 3][t];
        atomicAdd(&lhist[w][qbin(v.x)], 1u);
        atomicAdd(&lhist[w][qbin(v.y)], 1u);
        atomicAdd(&lhist[w][qbin(v.z)], 1u);
        atomicAdd(&lhist[w][qbin(v.w)], 1u);
    }
    __syncthreads();

    // reduce 8 sub-histograms, one global atomic per bin per block
    unsigned s = 0;
#pragma unroll
    for (int j = 0; j < 8; ++j) s += lhist[j][t];
    if (s) atomicAdd(&ghist[ch * NBINS + t], s);
}

// -------------------- pass 2: CDF -> LUT (one block per channel) --------------------
__global__ void __launch_bounds__(256) lut_kernel(const unsigned* __restrict__ ghist,
                                                  float* __restrict__ lut) {
    __shared__ unsigned scan[NBINS];
    __shared__ unsigned mred[NBINS];
    const int c = blockIdx.x;
    const int t = threadIdx.x;

    scan[t] = ghist[c * NBINS + t];
    __syncthreads();

    // Hillis-Steele inclusive scan over 256 bins (integer => exact, matches
    // the reference f32 cumsum since all partials <= 2^20 < 2^24).
#pragma unroll
    for (int off = 1; off < NBINS; off <<= 1) {
        unsigned v = (t >= off) ? scan[t - off] : 0u;
        __syncthreads();
        scan[t] += v;
        __syncthreads();
    }
    const unsigned cdf = scan[t];

    // cdf_min = min over cdf>0 (== first nonzero cdf, cdf nondecreasing)
    mred[t] = (cdf > 0u) ? cdf : 0xFFFFFFFFu;
    __syncthreads();
#pragma unroll
    for (int off = NBINS / 2; off > 0; off >>= 1) {
        if (t < off) {
            unsigned a = mred[t], b = mred[t + off];
            mred[t] = (a < b) ? a : b;
        }
        __syncthreads();
    }
    const float fm = (float)mred[0];
    const float denom = (float)NPIX - fm + 1e-8f;
    lut[c * NBINS + t] = ((float)cdf - fm) / denom;
}

// -------------------- pass 3: LUT remap --------------------
// 128 blocks per channel, 256 threads, 8 f4 per thread (8192 pixels/block).
// Loads re-hit L2 from pass 1 (192MB L2 vs 201MB image); stores are
// non-temporal so the output stream doesn't evict those lines. Loop is fully
// unrolled so all 8 b128 loads are in flight before the first LDS gather.
__global__ void __launch_bounds__(256) remap_kernel(const f4* __restrict__ img,
                                                    const float* __restrict__ lut,
                                                    f4* __restrict__ out) {
    __shared__ float slut[NBINS];
    const int t = threadIdx.x;
    const int ch  = blockIdx.x >> 7;       // 128 blocks per channel
    const int blk = blockIdx.x & 127;

    slut[t] = lut[ch * NBINS + t];
    __syncthreads();

    const size_t base = ((size_t)ch << 18) + ((size_t)blk * 2048);
#pragma unroll
    for (int k = 0; k < 8; ++k) {
        const size_t i = base + (size_t)k * 256 + t;
        f4 v = img[i];
        f4 o;
        o.x = slut[qbin(v.x)];
        o.y = slut[qbin(v.y)];
        o.z = slut[qbin(v.z)];
        o.w = slut[qbin(v.w)];
        __builtin_nontemporal_store(o, &out[i]);   // global_store_b128 th:NT
    }
}

// -------------------- launcher --------------------
extern "C" void kernel_launch(void* const* d_in, const int* in_sizes, int n_in,
                              void* d_out, int out_size, void* d_ws, size_t ws_size,
                              hipStream_t stream) {
    (void)in_sizes; (void)n_in; (void)out_size; (void)ws_size;

    const f4* img  = (const f4*)d_in[0];
    f4*       out  = (f4*)d_out;

    unsigned* ghist = (unsigned*)d_ws;                                  // 48*256 u32
    float*    lut   = (float*)((char*)d_ws + NCH * NBINS * sizeof(unsigned));

    hist_init   <<<NCH,       256, 0, stream>>>(ghist);
    hist_kernel <<<NCH * 64,  256, 0, stream>>>(img, ghist);
    lut_kernel  <<<NCH,       256, 0, stream>>>(ghist, lut);
    remap_kernel<<<NCH * 128, 256, 0, stream>>>(img, lut, out);
}